// WriteGCN_90391881711984
// MI455X (gfx1250) — compile-verified
//
#include <hip/hip_runtime.h>
#include <math.h>

namespace {

constexpr int   kAuthors = 100000;
constexpr int   kPapers  = 150000;
constexpr int   kDim     = 64;
constexpr int   kEdges   = 3200000;
constexpr int   kBatch   = 65536;
constexpr float kEps     = 1e-8f;

typedef __attribute__((ext_vector_type(2))) float v2f;
typedef __attribute__((ext_vector_type(8))) float v8f;

// ---------------------------------------------------------------------------
// 1) Degree histogram: deg_a[ea[e]] += 1, deg_p[ep[e]] += 1
// ---------------------------------------------------------------------------
__global__ void wgcn_degree(const int* __restrict__ ea, const int* __restrict__ ep,
                            float* __restrict__ deg_a, float* __restrict__ deg_p) {
  int e = blockIdx.x * blockDim.x + threadIdx.x;
  if (e >= kEdges) return;
  atomicAdd(&deg_a[ea[e]], 1.0f);
  atomicAdd(&deg_p[ep[e]], 1.0f);
}

// ---------------------------------------------------------------------------
// 2) Init layer accumulator: dst[i,:] = (1 - d/(d+eps)) * src[i,:]
//    (atomics from the edge pass then add directly on top -> no zero pass)
// ---------------------------------------------------------------------------
__global__ void wgcn_init_layer(const float* __restrict__ src,
                                const float* __restrict__ deg,
                                float* __restrict__ dst, int n_nodes) {
  int t = blockIdx.x * blockDim.x + threadIdx.x;   // one float4 per thread
  int total = n_nodes * (kDim / 4);
  if (t >= total) return;
  int node = t / (kDim / 4);
  float d  = deg[node];
  float sw = 1.0f - d / (d + kEps);                // ~0 unless isolated node
  float4 v = ((const float4*)src)[t];
  v.x *= sw; v.y *= sw; v.z *= sw; v.w *= sw;
  ((float4*)dst)[t] = v;
}

// ---------------------------------------------------------------------------
// 3) Edge pass: one wave32 per edge, each lane owns a float2 column pair.
//    p_dst[ep] += a_src[ea]*norm ; a_dst[ea] += p_src[ep]*norm
// ---------------------------------------------------------------------------
__global__ void wgcn_edge_pass(const int* __restrict__ ea_, const int* __restrict__ ep_,
                               const float* __restrict__ deg_a, const float* __restrict__ deg_p,
                               const float* __restrict__ a_src, const float* __restrict__ p_src,
                               float* __restrict__ a_dst, float* __restrict__ p_dst) {
  unsigned gt   = blockIdx.x * blockDim.x + threadIdx.x;
  unsigned e    = gt >> 5;           // wave -> edge (blockDim multiple of 32)
  unsigned lane = gt & 31;
  if (e >= kEdges) return;
  int a = ea_[e];
  int p = ep_[e];
  float norm = rsqrtf((deg_a[a] + kEps) * (deg_p[p] + kEps));
  unsigned col = lane * 2;
  size_t ao = (size_t)a * kDim + col;
  size_t po = (size_t)p * kDim + col;
  float2 am = *(const float2*)(a_src + ao);   // global_load_b64 (256B row / wave)
  float2 pm = *(const float2*)(p_src + po);
  atomicAdd(p_dst + po + 0, am.x * norm);
  atomicAdd(p_dst + po + 1, am.y * norm);
  atomicAdd(a_dst + ao + 0, pm.x * norm);
  atomicAdd(a_dst + ao + 1, pm.y * norm);
}

// ---------------------------------------------------------------------------
// 4) Final gather + outputs + WMMA row-dot + sigmoid.
//    One wave per 16 batch rows. Fragment mapping (V_WMMA_F32_16X16X4_F32):
//      A 16x4 f32 : lane<16  -> (M=lane,    K = k0+0 / k0+1) in a.x/a.y
//                   lane>=16 -> (M=lane-16, K = k0+2 / k0+3)
//      B 4x16 f32 : VGPR r: K=r (lanes 0-15), K=2+r (lanes 16-31), N=lane%16
//    With B[k][n] = LP[n][k] both fragments use col = k0 + 2*(lane>>4).
//    Diagonal of D: VGPR r = row r @ lane r, row 8+r @ lane 24+r.
// ---------------------------------------------------------------------------
__global__ void wgcn_predict(const float* __restrict__ ae, const float* __restrict__ pe,
                             const float* __restrict__ a1, const float* __restrict__ p1,
                             const float* __restrict__ a2, const float* __restrict__ p2,
                             const int* __restrict__ authors, const int* __restrict__ papers,
                             float* __restrict__ out) {
  unsigned gt   = blockIdx.x * blockDim.x + threadIdx.x;
  unsigned wave = gt >> 5;                 // 4096 waves, 16 rows each
  unsigned lane = gt & 31;
  unsigned row  = lane & 15;
  unsigned hi   = lane >> 4;
  unsigned brow = wave * 16 + row;
  if (brow >= kBatch) return;             // grid sized exactly; never diverges

  int ai = authors[brow];
  int pi = papers[brow];
  size_t abase = (size_t)ai * kDim;
  size_t pbase = (size_t)pi * kDim;
  float* out_la = out + kBatch;
  float* out_lp = out + kBatch + (size_t)kBatch * kDim;

  v8f c = {0.f, 0.f, 0.f, 0.f, 0.f, 0.f, 0.f, 0.f};
#pragma unroll
  for (int k0 = 0; k0 < kDim; k0 += 4) {
    int col = k0 + 2 * (int)hi;
    float2 la = *(const float2*)(ae + abase + col);
    float2 t  = *(const float2*)(a1 + abase + col); la.x += t.x; la.y += t.y;
    t         = *(const float2*)(a2 + abase + col); la.x += t.x; la.y += t.y;
    float2 lp = *(const float2*)(pe + pbase + col);
    t         = *(const float2*)(p1 + pbase + col); lp.x += t.x; lp.y += t.y;
    t         = *(const float2*)(p2 + pbase + col); lp.x += t.x; lp.y += t.y;
    // latest_author / latest_paper outputs reuse the same data the WMMA eats
    *(float2*)(out_la + (size_t)brow * kDim + col) = la;
    *(float2*)(out_lp + (size_t)brow * kDim + col) = lp;
    v2f af; af.x = la.x; af.y = la.y;
    v2f bf; bf.x = lp.x; bf.y = lp.y;
    c = __builtin_amdgcn_wmma_f32_16x16x4_f32(false, af, false, bf,
                                              (short)0, c, false, false);
  }

  // diagonal extraction + sigmoid
  bool active   = (lane < 8) || (lane >= 24);
  unsigned sel  = (lane < 8) ? lane : (lane - 24);
  unsigned mrow = (lane < 8) ? lane : (lane - 16);
  float diag = c[0];
  diag = (sel == 1) ? c[1] : diag;
  diag = (sel == 2) ? c[2] : diag;
  diag = (sel == 3) ? c[3] : diag;
  diag = (sel == 4) ? c[4] : diag;
  diag = (sel == 5) ? c[5] : diag;
  diag = (sel == 6) ? c[6] : diag;
  diag = (sel == 7) ? c[7] : diag;
  if (active) {
    out[wave * 16 + mrow] = 1.0f / (1.0f + expf(-diag));
  }
}

} // anonymous namespace

extern "C" void kernel_launch(void* const* d_in, const int* in_sizes, int n_in,
                              void* d_out, int out_size, void* d_ws, size_t ws_size,
                              hipStream_t stream) {
  (void)in_sizes; (void)n_in; (void)out_size; (void)ws_size;

  const float* ae      = (const float*)d_in[0];   // author_emb [100000,64]
  const float* pe      = (const float*)d_in[1];   // paper_emb  [150000,64]
  const int*   authors = (const int*)d_in[2];     // [65536]
  const int*   papers  = (const int*)d_in[3];     // [65536]
  const int*   ea      = (const int*)d_in[4];     // [3.2M]
  const int*   ep      = (const int*)d_in[5];     // [3.2M]
  float*       out     = (float*)d_out;           // [B] ++ [B,64] ++ [B,64]

  // ---- workspace carve-out (256B aligned), ~129 MB total ----
  char*  ws  = (char*)d_ws;
  size_t off = 0;
  auto take = [&](size_t bytes) -> void* {
    void* p = ws + off;
    off += (bytes + 255) & ~(size_t)255;
    return p;
  };
  float* deg_a = (float*)take((size_t)kAuthors * sizeof(float));
  float* deg_p = (float*)take((size_t)kPapers  * sizeof(float));
  float* a1    = (float*)take((size_t)kAuthors * kDim * sizeof(float));
  float* p1    = (float*)take((size_t)kPapers  * kDim * sizeof(float));
  float* a2    = (float*)take((size_t)kAuthors * kDim * sizeof(float));
  float* p2    = (float*)take((size_t)kPapers  * kDim * sizeof(float));

  // degrees
  hipMemsetAsync(deg_a, 0, (size_t)kAuthors * sizeof(float), stream);
  hipMemsetAsync(deg_p, 0, (size_t)kPapers  * sizeof(float), stream);
  wgcn_degree<<<(kEdges + 255) / 256, 256, 0, stream>>>(ea, ep, deg_a, deg_p);

  // layer 1: a1/p1 = sw*emb, then scatter-add messages
  wgcn_init_layer<<<(kAuthors * (kDim / 4) + 255) / 256, 256, 0, stream>>>(ae, deg_a, a1, kAuthors);
  wgcn_init_layer<<<(kPapers  * (kDim / 4) + 255) / 256, 256, 0, stream>>>(pe, deg_p, p1, kPapers);
  {
    size_t threads = (size_t)kEdges * 32;
    wgcn_edge_pass<<<(unsigned)((threads + 255) / 256), 256, 0, stream>>>(
        ea, ep, deg_a, deg_p, ae, pe, a1, p1);
  }

  // layer 2: a2/p2 = sw*l1, then scatter-add messages from l1
  wgcn_init_layer<<<(kAuthors * (kDim / 4) + 255) / 256, 256, 0, stream>>>(a1, deg_a, a2, kAuthors);
  wgcn_init_layer<<<(kPapers  * (kDim / 4) + 255) / 256, 256, 0, stream>>>(p1, deg_p, p2, kPapers);
  {
    size_t threads = (size_t)kEdges * 32;
    wgcn_edge_pass<<<(unsigned)((threads + 255) / 256), 256, 0, stream>>>(
        ea, ep, deg_a, deg_p, a1, p1, a2, p2);
  }

  // final: gather 3-layer sums, write latest_*, WMMA row-dot + sigmoid
  {
    unsigned waves   = kBatch / 16;              // 4096
    unsigned threads = waves * 32;               // 131072
    wgcn_predict<<<threads / 256, 256, 0, stream>>>(ae, pe, a1, p1, a2, p2,
                                                    authors, papers, out);
  }
}